// Encoder_20375324852398
// MI455X (gfx1250) — compile-verified
//
#include <hip/hip_runtime.h>

typedef float v2f __attribute__((ext_vector_type(2)));
typedef float v8f __attribute__((ext_vector_type(8)));

#define TILE_B   128
#define DDIM     64
#define A_STRIDE 68   // 16 consecutive rows -> 16 distinct bank groups (shift 4/row)

__global__ __launch_bounds__(256)
void encoder_wmma_kernel(const int*   __restrict__ nodes,
                         const float* __restrict__ emb,
                         const float* __restrict__ l1f,
                         const float* __restrict__ l2f,
                         const float* __restrict__ alpha_w,
                         const float* __restrict__ alpha_b,
                         const float* __restrict__ beta_w,
                         const float* __restrict__ beta_b,
                         const float* __restrict__ gamma_w,
                         const float* __restrict__ gamma_b,
                         float*       __restrict__ out)
{
    __shared__ float comb[TILE_B * A_STRIDE];        // combined tile (A-matrix source)
    __shared__ float bfrag[16 * 4 * 32 * 2];         // B fragments, per-lane packed:
                                                     // [ks][nt][lane][2] -> one b64/lane
    __shared__ float awS[2 * DDIM];
    __shared__ float bwS[2 * DDIM];
    __shared__ float gbS[DDIM];

    const int t = threadIdx.x;

    // ---- preload small weights into LDS ----
    if (t < 2 * DDIM) { awS[t] = alpha_w[t]; bwS[t] = beta_w[t]; }
    if (t < DDIM)     { gbS[t] = gamma_b[t]; }

    // gamma_w -> per-lane WMMA B-fragment layout (bijective scatter, 4096 elems)
    // element (ks, nt, lane, j): k = ks*4 + (lane>=16 ? 2 : 0) + j ; n = nt*16 + lane%16
    for (int idx = t; idx < 16 * 4 * 32 * 2; idx += 256) {
        const int j    =  idx        & 1;
        const int lane = (idx >> 1)  & 31;
        const int nt   = (idx >> 6)  & 3;
        const int ks   =  idx >> 8;
        const int k = ks * 4 + ((lane >> 4) << 1) + j;
        const int n = nt * 16 + (lane & 15);
        bfrag[idx] = gamma_w[n * DDIM + k];
    }
    __syncthreads();

    // ---- phase 1: gather + gates + combine (2 threads per row, 32 cols each) ----
    const int  row  = t >> 1;
    const int  half = t & 1;
    const int  c0   = half * 32;
    const long gRow = (long)blockIdx.x * TILE_B + row;

    const int node = nodes[gRow];
    const float* sp = emb + (long)node * DDIM + c0;
    const float* p1 = l1f + gRow * DDIM + c0;
    const float* p2 = l2f + gRow * DDIM + c0;

    float4 sv[8], v1[8], v2[8];
    float pa = 0.f, pb = 0.f;
#pragma unroll
    for (int j = 0; j < 8; ++j) {
        sv[j] = *(const float4*)(sp + 4 * j);
        v1[j] = *(const float4*)(p1 + 4 * j);
        v2[j] = *(const float4*)(p2 + 4 * j);
        const float* aw0 = awS + c0 + 4 * j;
        const float* aw1 = awS + DDIM + c0 + 4 * j;
        const float* bw0 = bwS + c0 + 4 * j;
        const float* bw1 = bwS + DDIM + c0 + 4 * j;
        pa += sv[j].x * aw0[0] + sv[j].y * aw0[1] + sv[j].z * aw0[2] + sv[j].w * aw0[3];
        pa += v1[j].x * aw1[0] + v1[j].y * aw1[1] + v1[j].z * aw1[2] + v1[j].w * aw1[3];
        pb += sv[j].x * bw0[0] + sv[j].y * bw0[1] + sv[j].z * bw0[2] + sv[j].w * bw0[3];
        pb += v2[j].x * bw1[0] + v2[j].y * bw1[1] + v2[j].z * bw1[2] + v2[j].w * bw1[3];
    }

    // partner thread (row-mate) is lane^1 in the same wave32
    float alpha = pa + __shfl_xor(pa, 1, 32) + alpha_b[0];
    float beta  = pb + __shfl_xor(pb, 1, 32) + beta_b[0];
    alpha = (alpha >= 0.f) ? alpha : 1.0f * alpha;   // PReLU(w=1.0)
    beta  = (beta  >= 0.f) ? beta  : 1.0f * beta;

    float* cr = comb + row * A_STRIDE + c0;
#pragma unroll
    for (int j = 0; j < 8; ++j) {
        float4 c;
        c.x = sv[j].x + alpha * v1[j].x + beta * v2[j].x;
        c.y = sv[j].y + alpha * v1[j].y + beta * v2[j].y;
        c.z = sv[j].z + alpha * v1[j].z + beta * v2[j].z;
        c.w = sv[j].w + alpha * v1[j].w + beta * v2[j].w;
        *(float4*)(cr + 4 * j) = c;
    }
    __syncthreads();

    // ---- phase 2: out = combined @ gamma_w^T + gamma_b via V_WMMA_F32_16X16X4_F32 ----
    const int wave = t >> 5;
    const int lane = t & 31;
    const int m    = lane & 15;               // A row in tile / N col for B,C,D
    const int hk   = ((lane >> 4) & 1) * 2;   // lane-half K offset (+2 for lanes 16-31)
    const int hi   = (lane >> 4) & 1;         // lane-half M offset for C/D (+8)
    const int r0   = wave * 16;

    v8f acc[4];
#pragma unroll
    for (int nt = 0; nt < 4; ++nt) {
        const float b = gbS[nt * 16 + m];     // bias folded into C
#pragma unroll
        for (int v = 0; v < 8; ++v) acc[nt][v] = b;
    }

    const float* arow = comb + (r0 + m) * A_STRIDE;
    const float* bf   = bfrag + lane * 2;     // this lane's packed fragment base
#pragma unroll
    for (int ks = 0; ks < 16; ++ks) {
        const int klo = ks * 4 + hk;
        const v2f a = *(const v2f*)(arow + klo);            // one b64, 8B aligned
#pragma unroll
        for (int nt = 0; nt < 4; ++nt) {
            const v2f b = *(const v2f*)(bf + (ks * 4 + nt) * 64);  // one b64/lane
            acc[nt] = __builtin_amdgcn_wmma_f32_16x16x4_f32(
                false, a, false, b, (short)0, acc[nt], false, false);
        }
    }

    // ---- store with PReLU (identity for w=1.0) ----
    const long outRow0 = (long)blockIdx.x * TILE_B + r0 + 8 * hi;
#pragma unroll
    for (int nt = 0; nt < 4; ++nt) {
#pragma unroll
        for (int v = 0; v < 8; ++v) {
            float x = acc[nt][v];
            x = (x >= 0.f) ? x : 1.0f * x;
            out[(outRow0 + v) * DDIM + nt * 16 + m] = x;
        }
    }
}

extern "C" void kernel_launch(void* const* d_in, const int* in_sizes, int n_in,
                              void* d_out, int out_size, void* d_ws, size_t ws_size,
                              hipStream_t stream) {
    const int*   nodes   = (const int*)  d_in[0];
    const float* emb     = (const float*)d_in[1];
    const float* l1f     = (const float*)d_in[2];
    const float* l2f     = (const float*)d_in[3];
    const float* alpha_w = (const float*)d_in[4];
    const float* alpha_b = (const float*)d_in[5];
    const float* beta_w  = (const float*)d_in[6];
    const float* beta_b  = (const float*)d_in[7];
    const float* gamma_w = (const float*)d_in[8];
    const float* gamma_b = (const float*)d_in[9];
    float* out = (float*)d_out;

    const int Bn   = in_sizes[0];          // 524288, divisible by 128
    const int grid = Bn / TILE_B;

    encoder_wmma_kernel<<<grid, 256, 0, stream>>>(
        nodes, emb, l1f, l2f, alpha_w, alpha_b, beta_w, beta_b,
        gamma_w, gamma_b, out);
}